// ModuleCorrelation_5128190951784
// MI455X (gfx1250) — compile-verified
//
#include <hip/hip_runtime.h>

typedef __attribute__((ext_vector_type(2))) float v2f;
typedef __attribute__((ext_vector_type(8))) float v8f;

namespace {
constexpr int Bn = 4;
constexpr int Cn = 256;
constexpr int Hn = 128;
constexpr int Wn = 224;
constexpr int Pn = 4;
constexpr int ND = 2 * Pn + 1;   // 9 displacements per axis
constexpr int NWAVES = 7;        // 7 waves x 2 adjacent x-tiles = 14 tiles
}

// One block handles (b, y, dy): the 9-wide diagonal band of the Gram matrix
// between first-row y and second-row z=y+dy-4, i.e. out[b, dy*9+dx, y, x].
// Wave w owns x-tiles 2w and 2w+1 (x in [32w, 32w+32)). Adjacent tiles share
// the middle A fragment, so 4 WMMAs need only 3 A + 2 B fragment loads.
__global__ __launch_bounds__(NWAVES * 32) void corr_wmma_kernel(
    const float* __restrict__ first,
    const float* __restrict__ second,
    float* __restrict__ out) {
  __shared__ float lds_out[ND * Wn];   // 9*224 floats = 8064 B

  const int blk = blockIdx.x;          // ((b*H + y)*9 + dy)
  const int dy  = blk % ND;
  const int by  = blk / ND;
  const int y   = by % Hn;
  const int b   = by / Hn;
  const int z   = y + dy - Pn;

  const int tid  = threadIdx.x;
  const int wave = tid >> 5;
  const int lane = tid & 31;
  const int half = lane >> 4;          // 0: lanes 0-15, 1: lanes 16-31
  const int ln   = lane & 15;          // M/N index within fragment

  const size_t cs = (size_t)Hn * Wn;   // channel stride (= dx-plane stride)
  float* outbase = out + (((size_t)(b * ND * ND + dy * ND)) * Hn + y) * Wn;

  if (z < 0 || z >= Hn) {
    // Entire dy slab is zero (padding region).
    for (int i = tid; i < ND * Wn; i += blockDim.x) {
      const int dx = i / Wn;
      const int x  = i - dx * Wn;
      outbase[(size_t)dx * cs + x] = 0.0f;
    }
    return;
  }

  const float* Frow = first  + ((size_t)(b * Cn) * Hn + y) * Wn;
  const float* Srow = second + ((size_t)(b * Cn) * Hn + z) * Wn;

  const int n0 = wave * 32;            // this wave's x base (two 16-tiles)

  // A (=S^T) fragment columns. A1 (n0+12..n0+27) is always in [12, 219]:
  // never needs masking. A0 can underflow (wave 0), A2 can overflow (wave 6).
  const int ca0 = n0 - 4  + ln;
  const int ca1 = n0 + 12 + ln;
  const int ca2 = n0 + 28 + ln;
  const int xb0 = n0 + ln;             // B fragment columns (always valid)
  const int xb1 = n0 + 16 + ln;

  // Branch-free boundary handling: clamp address, scale value by 0/1 mask.
  // EXEC stays all-ones through the K loop (WMMA requirement).
  const int   ca0c = ca0 < 0 ? 0 : ca0;
  const int   ca2c = ca2 >= Wn ? Wn - 1 : ca2;
  const float m0   = (ca0 >= 0) ? 1.0f : 0.0f;
  const float m2   = (ca2 < Wn) ? 1.0f : 0.0f;

  v8f d00 = {};   // A0 x B0 : tile 2w,   dx = m - n
  v8f d01 = {};   // A1 x B0 : tile 2w,   dx = m + 16 - n
  v8f d11 = {};   // A1 x B1 : tile 2w+1, dx = m - n
  v8f d21 = {};   // A2 x B1 : tile 2w+1, dx = m + 16 - n

#pragma unroll 4
  for (int k = 0; k < Cn; k += 4) {
    // 32-bit A 16x4 / B 4x16 layout: lane half selects K pair, K = 2*half + j.
    const int c0 = k + 2 * half;
    const float* Fp = Frow + (size_t)c0 * cs;
    const float* Sp = Srow + (size_t)c0 * cs;

    v2f b0;  b0.x = Fp[xb0];       b0.y = Fp[cs + xb0];
    v2f b1;  b1.x = Fp[xb1];       b1.y = Fp[cs + xb1];
    v2f a0;  a0.x = Sp[ca0c] * m0; a0.y = Sp[cs + ca0c] * m0;
    v2f a1;  a1.x = Sp[ca1];       a1.y = Sp[cs + ca1];
    v2f a2;  a2.x = Sp[ca2c] * m2; a2.y = Sp[cs + ca2c] * m2;

    d00 = __builtin_amdgcn_wmma_f32_16x16x4_f32(
        false, a0, false, b0, (short)0, d00, false, false);
    d01 = __builtin_amdgcn_wmma_f32_16x16x4_f32(
        false, a1, false, b0, (short)0, d01, false, false);
    d11 = __builtin_amdgcn_wmma_f32_16x16x4_f32(
        false, a1, false, b1, (short)0, d11, false, false);
    d21 = __builtin_amdgcn_wmma_f32_16x16x4_f32(
        false, a2, false, b1, (short)0, d21, false, false);
  }

  // D layout: lane holds N=ln, M = v + 8*half (upper A tile: +16).
  // Band entry: dx = m_abs - n in [0, 9). Each (dx, x) lands in exactly one
  // (acc, half, v) slot, so every output is written exactly once.
#pragma unroll
  for (int v = 0; v < 8; ++v) {
    const int m   = v + 8 * half;
    const int dxL = m - ln;          // from d00 (x tile 2w) / d11 (tile 2w+1)
    const int dxH = m + 16 - ln;     // from d01 (x tile 2w) / d21 (tile 2w+1)
    if (dxL >= 0 && dxL < ND) {
      lds_out[dxL * Wn + n0 + ln]      = d00[v];
      lds_out[dxL * Wn + n0 + 16 + ln] = d11[v];
    }
    if (dxH >= 0 && dxH < ND) {
      lds_out[dxH * Wn + n0 + ln]      = d01[v];
      lds_out[dxH * Wn + n0 + 16 + ln] = d21[v];
    }
  }

  __syncthreads();

  // Coalesced store of the 9 dx-rows, scaled by 1/C.
  for (int i = tid; i < ND * Wn; i += blockDim.x) {
    const int dx = i / Wn;
    const int x  = i - dx * Wn;
    outbase[(size_t)dx * cs + x] = lds_out[i] * (1.0f / (float)Cn);
  }
}

extern "C" void kernel_launch(void* const* d_in, const int* in_sizes, int n_in,
                              void* d_out, int out_size, void* d_ws, size_t ws_size,
                              hipStream_t stream) {
  const float* first  = (const float*)d_in[0];
  const float* second = (const float*)d_in[1];
  float* out = (float*)d_out;

  const int nblocks = Bn * Hn * ND;    // 4608 blocks of 224 threads (7 waves)
  corr_wmma_kernel<<<nblocks, NWAVES * 32, 0, stream>>>(first, second, out);
}